// NoiseMemoryBank_5248450035798
// MI455X (gfx1250) — compile-verified
//
#include <hip/hip_runtime.h>

// Gather: out[b, :] = bank[cid[b], ridx[b], :]   (768 f32 per row, 65536 rows)
// Pure streaming copy: ~402 MB total traffic -> ~17us floor @ 23.3 TB/s.
// Full-block kernel: 192 threads/block, 8 rows/block, gfx1250 async
// global->LDS->global DMA path (ASYNCcnt). Indices fetched as one s_load_b256
// per array; all addressing in u32 byte offsets (bank = 1.61GB < 2^31).
// Tail rows (batch % 8) handled by a separate trivial kernel.

#define FEAT     768
#define CAP      2048
#define RPB      8                       // rows per block (full kernel)
#define THREADS  192                     // FEAT*4 bytes / 16 bytes-per-lane
#define ROWBYTES (FEAT * 4u)             // 3072
#define CTRBYTES (CAP * FEAT * 4u)       // 6291456

typedef __attribute__((ext_vector_type(4))) float v4f;
typedef __attribute__((ext_vector_type(4))) int   v4i;
typedef __attribute__((ext_vector_type(8))) int   v8i;

#ifndef __has_builtin
#define __has_builtin(x) 0
#endif

#if __has_builtin(__builtin_amdgcn_global_load_async_to_lds_b128) && \
    __has_builtin(__builtin_amdgcn_global_store_async_from_lds_b128)
#define HAVE_ASYNC_LDS 1
#else
#define HAVE_ASYNC_LDS 0
#endif

#if __has_builtin(__builtin_amdgcn_s_wait_asynccnt)
#define WAIT_ASYNC0() __builtin_amdgcn_s_wait_asynccnt(0)
#else
#define WAIT_ASYNC0() asm volatile("s_wait_asynccnt 0" ::: "memory")
#endif

// Pointer types per the builtin signature:
// (v4i addrspace(1)*, v4i addrspace(3)*, imm offset, imm cpol)
typedef __attribute__((address_space(1))) v4i* g_v4i_p;
typedef __attribute__((address_space(3))) v4i* l_v4i_p;

// ---------------- full blocks: no bounds checks, no clamps ----------------
__global__ __launch_bounds__(THREADS) void gather_rows_full(
    const float* __restrict__ bank,
    const int*   __restrict__ cid,
    const int*   __restrict__ ridx,
    float*       __restrict__ out)
{
    const int      t    = threadIdx.x;            // 0..191
    const unsigned row0 = blockIdx.x * RPB;       // multiple of 8 -> 32B aligned idx
    const unsigned tb   = (unsigned)t * 16u;      // lane byte offset within row

    // One 32-byte uniform vector load per index array -> s_load_b256 each.
    const v8i c = *(const v8i*)(cid  + row0);
    const v8i k = *(const v8i*)(ridx + row0);

#if HAVE_ASYNC_LDS
    // CDNA5 async path: HBM -> LDS -> HBM without touching VGPRs for the data.
    __shared__ __align__(16) char lds[RPB * FEAT * sizeof(float)];

    #pragma unroll
    for (int r = 0; r < RPB; ++r) {
        // u32 byte offset: max 255*6291456 + 2047*3072 + 3056 < 2^31
        const unsigned off = (unsigned)c[r] * CTRBYTES + (unsigned)k[r] * ROWBYTES + tb;
        __builtin_amdgcn_global_load_async_to_lds_b128(
            (g_v4i_p)(v4i*)(void*)((char*)bank + off),
            (l_v4i_p)(v4i*)(void*)(lds + (unsigned)r * (FEAT * 4u) + tb),
            0, 0);
    }
    // Each lane stores exactly the LDS bytes it loaded -> only the ASYNCcnt
    // wait is needed (LDS side of async ops is otherwise unordered).
    WAIT_ASYNC0();

    #pragma unroll
    for (int r = 0; r < RPB; ++r) {
        const unsigned off = (row0 + (unsigned)r) * ROWBYTES + tb;   // < 2^31
        __builtin_amdgcn_global_store_async_from_lds_b128(
            (g_v4i_p)(v4i*)(void*)((char*)out + off),
            (l_v4i_p)(v4i*)(void*)(lds + (unsigned)r * (FEAT * 4u) + tb),
            0, 0);
    }
    // S_ENDPGM performs an implicit wait-idle, draining outstanding async stores.
#else
    v4f v[RPB];
    #pragma unroll
    for (int r = 0; r < RPB; ++r) {
        const unsigned off = (unsigned)c[r] * CTRBYTES + (unsigned)k[r] * ROWBYTES + tb;
        v[r] = __builtin_nontemporal_load((const v4f*)((const char*)bank + off));
    }
    #pragma unroll
    for (int r = 0; r < RPB; ++r) {
        const unsigned off = (row0 + (unsigned)r) * ROWBYTES + tb;
        __builtin_nontemporal_store(v[r], (v4f*)((char*)out + off));
    }
#endif
}

// ---------------- tail: one row per block, simple direct copy ----------------
__global__ __launch_bounds__(THREADS) void gather_rows_tail(
    const float* __restrict__ bank,
    const int*   __restrict__ cid,
    const int*   __restrict__ ridx,
    float*       __restrict__ out,
    int start, int batch)
{
    const int b = start + (int)blockIdx.x;
    if (b >= batch) return;
    const int t = threadIdx.x;
    const unsigned off = (unsigned)cid[b] * CTRBYTES + (unsigned)ridx[b] * ROWBYTES
                       + (unsigned)t * 16u;
    const v4f v = __builtin_nontemporal_load((const v4f*)((const char*)bank + off));
    __builtin_nontemporal_store(
        v, (v4f*)((char*)out + (unsigned)b * ROWBYTES + (unsigned)t * 16u));
}

extern "C" void kernel_launch(void* const* d_in, const int* in_sizes, int n_in,
                              void* d_out, int out_size, void* d_ws, size_t ws_size,
                              hipStream_t stream) {
    const float* bank = (const float*)d_in[0];
    const int*   cid  = (const int*)d_in[1];   // target_center_ids
    const int*   ridx = (const int*)d_in[2];   // rand_idx
    float*       out  = (float*)d_out;

    const int batch = in_sizes[1];             // 65536
    const int nfull = batch / RPB;             // 8192 full blocks
    const int rem   = batch % RPB;             // 0 for this shape

    if (nfull > 0)
        gather_rows_full<<<nfull, THREADS, 0, stream>>>(bank, cid, ridx, out);
    if (rem > 0)
        gather_rows_tail<<<rem, THREADS, 0, stream>>>(bank, cid, ridx, out,
                                                      nfull * RPB, batch);
}